// SpanQuery_5995774345592
// MI455X (gfx1250) — compile-verified
//
#include <hip/hip_runtime.h>

// CDNA5 / gfx1250: wave32, WMMA 16x16 tiles.
typedef __attribute__((ext_vector_type(2))) float v2f;
typedef __attribute__((ext_vector_type(8))) float v8f;

#define HIDDEN 768
#define SPAN   12
#define MTOT   4096   // 8 * 512 flattened (b,l)

// Each workgroup: 4 waves, one 16-row m-tile of h staged in LDS (48 KB),
// each wave owns a 16-wide e-subtile and all 12 span accumulators.
__global__ __launch_bounds__(128)
void spanquery_wmma_kernel(const float* __restrict__ h,
                           const float* __restrict__ qs,   // (768, 12) row-major
                           const float* __restrict__ W,    // (768, 768) row-major: W[e][d]
                           const float* __restrict__ bias, // (768)
                           float* __restrict__ out)        // (4096, 12, 768)
{
    __shared__ float hTile[16 * HIDDEN];   // 48 KB

    const int tid  = threadIdx.x;          // 0..127
    const int wave = tid >> 5;             // 0..3
    const int lane = tid & 31;
    const int half = lane >> 4;            // 0: lanes 0-15, 1: lanes 16-31
    const int l16  = lane & 15;

    const int mBase = blockIdx.x * 16;                   // m-tile origin
    const int e     = blockIdx.y * 64 + wave * 16 + l16; // output feature (B column)

    // --- Cooperative stage of h tile: 16 rows x 768 cols, fully coalesced b128 ---
    {
        const float4* src = (const float4*)(h + (size_t)mBase * HIDDEN);
        float4*       dst = (float4*)hTile;
        #pragma unroll
        for (int i = 0; i < (16 * HIDDEN / 4) / 128; ++i)   // 24 float4 per thread
            dst[tid + i * 128] = src[tid + i * 128];
    }
    __syncthreads();

    // f32 WMMA 16x16x4 operand layout (ISA 7.12.2):
    //   A: lane holds M = lane%16; VGPR0/1 = K = {0,1} (lanes 0-15) or {2,3} (lanes 16-31)
    //   B: lane holds N = lane%16; same K split per half-wave
    const int koff = half * 2;
    const float* Wrow = W + (size_t)e * HIDDEN;

    v8f acc[SPAN] = {};   // 12 x (16x16 f32) accumulators = 96 VGPRs

    for (int k = 0; k < HIDDEN; k += 4) {
        const int kk = k + koff;

        // A raw pair from LDS (8B aligned: kk is even)
        const float2 hp = *(const float2*)&hTile[l16 * HIDDEN + kk];

        // B pair: W[e][kk], W[e][kk+1]  (shared across all 12 spans)
        v2f bvec;
        {
            const float2 wp = *(const float2*)&Wrow[kk];
            bvec.x = wp.x; bvec.y = wp.y;
        }

        // query_seg rows kk and kk+1 (12 floats each; (kk*12*4)B is 16B aligned)
        float q0r[SPAN], q1r[SPAN];
        {
            const float4* q0v = (const float4*)(qs + kk * SPAN);
            const float4* q1v = (const float4*)(qs + (kk + 1) * SPAN);
            ((float4*)q0r)[0] = q0v[0]; ((float4*)q0r)[1] = q0v[1]; ((float4*)q0r)[2] = q0v[2];
            ((float4*)q1r)[0] = q1v[0]; ((float4*)q1r)[1] = q1v[1]; ((float4*)q1r)[2] = q1v[2];
        }

        // 12 WMMAs per K-step: A_s = hp * qs (folded), B shared.
        #pragma unroll
        for (int s = 0; s < SPAN; ++s) {
            v2f avec;
            avec.x = hp.x * q0r[s];
            avec.y = hp.y * q1r[s];
            acc[s] = __builtin_amdgcn_wmma_f32_16x16x4_f32(
                /*neg_a=*/false, avec, /*neg_b=*/false, bvec,
                /*c_mod=*/(short)0, acc[s], /*reuse_a=*/false, /*reuse_b=*/false);
        }
    }

    // --- Epilogue: bias + ReLU, store. C/D layout: lane%16 = N(e), VGPR v = M row,
    //     lanes 16-31 carry M+8. Each half-wave writes 64 contiguous bytes. ---
    const float bv = bias[e];
    #pragma unroll
    for (int s = 0; s < SPAN; ++s) {
        #pragma unroll
        for (int v = 0; v < 8; ++v) {
            const int m = mBase + v + half * 8;
            float r = acc[s][v] + bv;
            r = fmaxf(r, 0.0f);
            out[((size_t)m * SPAN + s) * HIDDEN + e] = r;
        }
    }
}

extern "C" void kernel_launch(void* const* d_in, const int* in_sizes, int n_in,
                              void* d_out, int out_size, void* d_ws, size_t ws_size,
                              hipStream_t stream) {
    (void)in_sizes; (void)n_in; (void)out_size; (void)d_ws; (void)ws_size;
    const float* h  = (const float*)d_in[0];   // (8,512,768) f32
    const float* qs = (const float*)d_in[1];   // (768,12)    f32
    const float* W  = (const float*)d_in[2];   // (768,768)   f32
    const float* b  = (const float*)d_in[3];   // (768)       f32
    float* out = (float*)d_out;                // (8,512,12,768) f32

    dim3 grid(MTOT / 16, HIDDEN / 64, 1);      // (256, 12)
    dim3 block(128, 1, 1);                     // 4 waves
    hipLaunchKernelGGL(spanquery_wmma_kernel, grid, block, 0, stream,
                       h, qs, W, b, out);
}